// SegNet_pool_layer_36807869726730
// MI455X (gfx1250) — compile-verified
//
#include <hip/hip_runtime.h>
#include <hip/hip_bf16.h>
#include <stdint.h>

#define NUM_NODES 40962
#define FEAT 256
#define NEIGH 7

// One block of 256 threads (8 wave32) per node.
// Waves 0..6 async-copy one gathered row each (1 KB) into LDS via the CDNA5
// async global->LDS path; then each thread does a 7-way max/argmax over its
// stride-7 slice of the concatenated 1792-float buffer.
__global__ __launch_bounds__(256) void SegNet_pool_layer_kernel(
    const float* __restrict__ x,        // (163842, 256) f32
    const int*   __restrict__ neigh,    // (40962*7,) int32 row indices
    float* __restrict__ out_vals,       // (40962, 256) f32
    float* __restrict__ out_idx)        // (40962, 256) argmax as f32
{
    __shared__ float cbuf[FEAT * NEIGH];   // 1792 floats = 7 KB

    const int n    = blockIdx.x;
    const int t    = threadIdx.x;
    const int wave = t >> 5;               // 0..7
    const int lane = t & 31;

    // Hardware LDS byte address of cbuf: generic pointer to an LDS location is
    // {aperture_hi32, lds_offset32}, so the low 32 bits ARE the DS address.
    // The ptrtoint also captures cbuf, so the "memory"-clobbered asm below is
    // correctly assumed to write it (keeps the later DS reads alive).
    const unsigned lds_base = (unsigned)(uintptr_t)(void*)cbuf;

    if (wave < NEIGH) {
        const int row = neigh[n * NEIGH + wave];
        // Global byte address this lane copies from (16 B per lane).
        const unsigned long long gaddr =
            (unsigned long long)(uintptr_t)(x + (size_t)row * FEAT)
            + (unsigned)(lane * 16);
        // LDS byte address: rows packed contiguously -> cbuf[p] == c[p].
        const unsigned ldsaddr = lds_base + (unsigned)(wave * (FEAT * 4) + lane * 16);

        // 32 lanes x 16 B = first 512 B of the row.
        asm volatile("global_load_async_to_lds_b128 %0, %1, off"
                     :: "v"(ldsaddr), "v"(gaddr) : "memory");
        // IOFFSET applies to BOTH the LDS and global addresses -> second 512 B.
        asm volatile("global_load_async_to_lds_b128 %0, %1, off offset:512"
                     :: "v"(ldsaddr), "v"(gaddr) : "memory");
    }
    // Each wave drains its own async copies before the workgroup barrier.
    asm volatile("s_wait_asynccnt 0x0" ::: "memory");
    __syncthreads();

    // c[p] = x[neigh[7n + p/256], p%256]; vals[n,f] = max_k c[7f+k].
    const int base = t * NEIGH;            // stride 7 (odd) -> bank-conflict free
    float best = cbuf[base];
    int   bi   = 0;
#pragma unroll
    for (int k = 1; k < NEIGH; ++k) {
        const float v = cbuf[base + k];
        if (v > best) { best = v; bi = k; }   // strict '>' keeps first max (jnp.argmax)
    }

    const size_t o = (size_t)n * FEAT + t;
    out_vals[o] = best;
    out_idx[o]  = (float)bi;
}

extern "C" void kernel_launch(void* const* d_in, const int* in_sizes, int n_in,
                              void* d_out, int out_size, void* d_ws, size_t ws_size,
                              hipStream_t stream) {
    (void)in_sizes; (void)n_in; (void)out_size; (void)d_ws; (void)ws_size;
    const float* x     = (const float*)d_in[0];
    const int*   neigh = (const int*)d_in[1];
    float* out_vals = (float*)d_out;
    float* out_idx  = out_vals + (size_t)NUM_NODES * FEAT;

    SegNet_pool_layer_kernel<<<NUM_NODES, 256, 0, stream>>>(x, neigh, out_vals, out_idx);
}